// Model_11888469476112
// MI455X (gfx1250) — compile-verified
//
#include <hip/hip_runtime.h>

typedef __attribute__((ext_vector_type(2))) float v2f;
typedef __attribute__((ext_vector_type(4))) float v4f;
typedef __attribute__((ext_vector_type(8))) float v8f;
typedef __attribute__((ext_vector_type(2))) int   v2i;

#define CH_TOTAL      256        // B*C = 4*64
#define POOLED_PER_CH 32768      // 32^3
#define S_VOL         262144     // 64^3
#define INV_S         (1.0f / 262144.0f)

// ---------------------------------------------------------------------------
// Kernel 1: per-channel  gate = log_sigmoid( sum(silu(x)) / 64^3 )
// One block per (b,c) channel; 256 threads; 32 MB total read.
// Cross-lane reduction uses V_WMMA_F32_16X16X4_F32 with B = ones:
//   A VGPR0 = lane partial p, VGPR1 = 0  =>  D[m,n] = p_m + p_{m+16}
// After the horizontal add of the 8 C/D VGPRs, lanes 0-15 hold sum(p_0..7,
// p_16..23) and lanes 16-31 hold sum(p_8..15, p_24..31): lanes 0 and 16 of
// each wave write their half-sums to LDS (no shuffle needed).
// ---------------------------------------------------------------------------
__global__ __launch_bounds__(256) void gate_kernel(const float* __restrict__ pooled,
                                                   float* __restrict__ gates) {
  const int ch  = blockIdx.x;
  const int tid = threadIdx.x;
  const float* p = pooled + (size_t)ch * POOLED_PER_CH;

  // 32768 floats / 256 threads = 128 per thread, as 32 x float4 (b128 loads)
  float a0 = 0.0f, a1 = 0.0f, a2 = 0.0f, a3 = 0.0f;
  #pragma unroll 8
  for (int i = 0; i < 32; ++i) {
    v4f x = *(const v4f*)(p + i * 1024 + tid * 4);
    // silu(v) = v * sigmoid(v); v_rcp_f32 instead of full IEEE divide
    a0 += x[0] * __builtin_amdgcn_rcpf(1.0f + __expf(-x[0]));
    a1 += x[1] * __builtin_amdgcn_rcpf(1.0f + __expf(-x[1]));
    a2 += x[2] * __builtin_amdgcn_rcpf(1.0f + __expf(-x[2]));
    a3 += x[3] * __builtin_amdgcn_rcpf(1.0f + __expf(-x[3]));
  }
  const float acc = (a0 + a1) + (a2 + a3);

  // Wave-level reduce via WMMA (EXEC all-ones here: full blocks, no divergence)
  v2f a; a.x = acc;  a.y = 0.0f;
  v2f b; b.x = 1.0f; b.y = 1.0f;
  v8f c = {};
  c = __builtin_amdgcn_wmma_f32_16x16x4_f32(false, a, false, b, (short)0, c,
                                            false, false);
  const float half32 =
      ((c[0] + c[1]) + (c[2] + c[3])) + ((c[4] + c[5]) + (c[6] + c[7]));

  // 8 waves x 2 half-sums -> 16 LDS partials (lanes 0 and 16 of each wave)
  __shared__ float lds[16];
  if ((tid & 15) == 0) lds[tid >> 4] = half32;
  __syncthreads();

  if (tid == 0) {
    float s = 0.0f;
    #pragma unroll
    for (int w = 0; w < 16; ++w) s += lds[w];
    const float m = s * INV_S;
    // numerically stable log_sigmoid
    const float g = (m >= 0.0f) ? -log1pf(__expf(-m))
                                : (m - log1pf(__expf(m)));
    gates[ch] = g;
  }
}

// ---------------------------------------------------------------------------
// Kernel 2: gather-form MaxUnpool3d * gate.
// Thread owns the window pair (wp = 2j, 2j+1) at (ch, dp, hp):
//   loads int2 indices (NT, read-once) + float2 values (coalesced b64),
//   writes 4 output rows of 4 consecutive floats each (NT b128 stores).
// Lanes 0..15 of a wave cover one contiguous 256B row segment per store.
// Every output element is written exactly once -> no zero-fill pass needed.
// ch is block-uniform (64 blocks per channel) -> gate load & bases scalarize.
// ---------------------------------------------------------------------------
__global__ __launch_bounds__(256) void unpool_kernel(const float* __restrict__ pooled,
                                                     const int*   __restrict__ indices,
                                                     const float* __restrict__ gates,
                                                     float*       __restrict__ out) {
  const int ch = blockIdx.x >> 6;                       // 0..255, block-uniform
  const int w  = (blockIdx.x & 63) * 256 + threadIdx.x; // window pair in channel
  const int j  = w & 15;                                // wp = 2j, 2j+1
  const int hp = (w >> 4) & 31;
  const int dp = w >> 9;

  const int in_off = ch * POOLED_PER_CH + dp * 1024 + hp * 32 + 2 * j;
  const v2i idx2 = __builtin_nontemporal_load((const v2i*)(indices + in_off));
  const v2f val2 = *(const v2f*)(pooled + in_off);

  const float g  = gates[ch];
  const float gA = g * val2.x;    // window A (wp = 2j)
  const float gB = g * val2.y;    // window B (wp = 2j+1)
  const int   iA = idx2.x;
  const int   iB = idx2.y;

  float* o = out + (size_t)ch * S_VOL;
  // Output w-base for this pair: w0 = 4j; window A covers w0,w0+1; B covers w0+2,w0+3
  const int base = ((2 * dp) * 64 + 2 * hp) * 64 + 4 * j;

  #pragma unroll
  for (int d = 0; d < 2; ++d) {
    #pragma unroll
    for (int h = 0; h < 2; ++h) {
      const int r = base + d * 4096 + h * 64;   // flat offset of this row's 4 floats
      v4f v;
      v.x = (iA == r    ) ? gA : 0.0f;
      v.y = (iA == r + 1) ? gA : 0.0f;
      v.z = (iB == r + 2) ? gB : 0.0f;
      v.w = (iB == r + 3) ? gB : 0.0f;
      __builtin_nontemporal_store(v, (v4f*)(o + r));  // 256MB stream: bypass L2 retention
    }
  }
}

extern "C" void kernel_launch(void* const* d_in, const int* in_sizes, int n_in,
                              void* d_out, int out_size, void* d_ws, size_t ws_size,
                              hipStream_t stream) {
  const float* pooled  = (const float*)d_in[0];
  const int*   indices = (const int*)  d_in[1];
  float*       gates   = (float*)d_ws;           // 256 floats of scratch
  float*       out     = (float*)d_out;

  gate_kernel  <<<CH_TOTAL, 256, 0, stream>>>(pooled, gates);
  unpool_kernel<<<16384,    256, 0, stream>>>(pooled, indices, gates, out);

  (void)in_sizes; (void)n_in; (void)out_size; (void)ws_size;
}